// AtomEncoder_1898375545266
// MI455X (gfx1250) — compile-verified
//
#include <hip/hip_runtime.h>
#include <hip/hip_bf16.h>

typedef float v2f __attribute__((ext_vector_type(2)));
typedef float v8f __attribute__((ext_vector_type(8)));

// D = A(16x4) * B(4x16) + C, fp32 — V_WMMA_F32_16X16X4_F32
#define WMMA_F32(a, b, c) \
  __builtin_amdgcn_wmma_f32_16x16x4_f32(false, (a), false, (b), (short)0, (c), false, false)

// Async global->LDS staging (ASYNCcnt path), guarded so compile never breaks.
#if __has_builtin(__builtin_amdgcn_global_load_async_to_lds_b128) && \
    __has_builtin(__builtin_amdgcn_s_wait_asynccnt)
#define USE_ASYNC_LDS 1
#else
#define USE_ASYNC_LDS 0
#endif

constexpr int N_ATOM = 4096;
constexpr int T_TOK  = 512;
constexpr int ATOM_S = 128;
constexpr int TOK_S  = 384;
constexpr int TOK_Z  = 128;
constexpr int FEAT   = 388;

// workspace layout (floats)
constexpr int WS_SMEAN = 0;                    // 512
constexpr int WS_SRSTD = 512;                  // 512
constexpr int WS_S2C   = 1024;                 // 512*128
constexpr int WS_ZP    = WS_S2C + 512 * 128;   // 128*4*16*16 = 131072
constexpr int WS_PQ    = WS_ZP + 131072;       // 4096*16
constexpr int WS_PK    = WS_PQ + 65536;        // 4096*16

// ---------------------------------------------------------------- kernel A
__device__ __forceinline__ float atom_feat(const float* pos, const float* chg,
                                           const float* elem, const float* chars,
                                           int n, int f) {
  if (f < 3)   return pos[n * 3 + f];
  if (f == 3)  return chg[n];
  if (f < 132) return elem[n * 128 + (f - 4)];
  return chars[n * 256 + (f - 132)];
}

// q = atom_feats @ W_atom_feat^T + b   [4096 x 388] x [388 x 128]
__global__ void k_atom_embed(const float* __restrict__ pos, const float* __restrict__ chg,
                             const float* __restrict__ elem, const float* __restrict__ chars,
                             const float* __restrict__ Wf, const float* __restrict__ bf,
                             float* __restrict__ q_out) {
  int wv   = (blockIdx.x * blockDim.x + threadIdx.x) >> 5;  // M tile id (0..255)
  int lane = threadIdx.x & 31;
  int n16  = lane & 15;
  int half = lane >> 4;
  int m    = wv * 16 + n16;
  v8f acc[8] = {};
  for (int k0 = 0; k0 < FEAT; k0 += 4) {
    int kk = k0 + 2 * half;
    v2f a;
    a.x = atom_feat(pos, chg, elem, chars, m, kk);
    a.y = atom_feat(pos, chg, elem, chars, m, kk + 1);
#pragma unroll
    for (int nt = 0; nt < 8; nt++) {
      int n = nt * 16 + n16;
      v2f b;
      b.x = Wf[n * FEAT + kk];
      b.y = Wf[n * FEAT + kk + 1];
      acc[nt] = WMMA_F32(a, b, acc[nt]);
    }
  }
#pragma unroll
  for (int nt = 0; nt < 8; nt++) {
    int n  = nt * 16 + n16;
    float bb = bf[n];
#pragma unroll
    for (int i = 0; i < 8; i++) {
      int row = wv * 16 + i + 8 * half;
      q_out[row * ATOM_S + n] = acc[nt][i] + bb;
    }
  }
}

// ---------------------------------------------------------------- kernel B1
__global__ void k_s_stats(const float* __restrict__ s, float* __restrict__ mean,
                          float* __restrict__ rstd) {
  int wv   = (blockIdx.x * blockDim.x + threadIdx.x) >> 5;  // row 0..511
  int lane = threadIdx.x & 31;
  const float* row = s + wv * TOK_S;
  float s1 = 0.f, s2 = 0.f;
  for (int i = lane; i < TOK_S; i += 32) { float x = row[i]; s1 += x; s2 += x * x; }
  for (int off = 16; off; off >>= 1) { s1 += __shfl_xor(s1, off, 32); s2 += __shfl_xor(s2, off, 32); }
  if (lane == 0) {
    float m = s1 / TOK_S;
    mean[wv] = m;
    rstd[wv] = rsqrtf(s2 / TOK_S - m * m + 1e-5f);
  }
}

// ---------------------------------------------------------------- kernel B2
// s_to_c = LN(s_trunk) @ W_s_to_c^T   [512 x 384] x [384 x 128]
__global__ void k_s_to_c(const float* __restrict__ s, const float* __restrict__ mean,
                         const float* __restrict__ rstd, const float* __restrict__ lnsc,
                         const float* __restrict__ lnsb, const float* __restrict__ Wsc,
                         float* __restrict__ s2c) {
  int wv   = (blockIdx.x * blockDim.x + threadIdx.x) >> 5;  // M tile 0..31
  int lane = threadIdx.x & 31;
  int n16  = lane & 15;
  int half = lane >> 4;
  int m    = wv * 16 + n16;
  float mu = mean[m], rs = rstd[m];
  v8f acc[8] = {};
  for (int k0 = 0; k0 < TOK_S; k0 += 4) {
    int kk = k0 + 2 * half;
    v2f a;
    a.x = (s[m * TOK_S + kk]     - mu) * rs * lnsc[kk]     + lnsb[kk];
    a.y = (s[m * TOK_S + kk + 1] - mu) * rs * lnsc[kk + 1] + lnsb[kk + 1];
#pragma unroll
    for (int nt = 0; nt < 8; nt++) {
      int n = nt * 16 + n16;
      v2f b;
      b.x = Wsc[n * TOK_S + kk];
      b.y = Wsc[n * TOK_S + kk + 1];
      acc[nt] = WMMA_F32(a, b, acc[nt]);
    }
  }
#pragma unroll
  for (int nt = 0; nt < 8; nt++) {
    int n = nt * 16 + n16;
#pragma unroll
    for (int i = 0; i < 8; i++) {
      int row = wv * 16 + i + 8 * half;
      s2c[row * ATOM_S + n] = acc[nt][i];
    }
  }
}

// ---------------------------------------------------------------- kernel B3
// c = q + s_to_c[uid[n]]   (atom_to_token is one-hot(uid) -> gather)
__global__ void k_c_add(const float* __restrict__ qv, const float* __restrict__ s2c,
                        const int* __restrict__ uid, float* __restrict__ cv) {
  int idx = blockIdx.x * blockDim.x + threadIdx.x;
  int n = idx >> 7, ch = idx & 127;
  cv[idx] = qv[idx] + s2c[uid[n] * ATOM_S + ch];
}

// ---------------------------------------------------------------- kernel Z
// Compact z_to_p table: only 8192 (tq,tk) pairs are ever used by the local
// window attention. Wave per pair: LN over 128 channels + 128x16 projection.
__global__ void k_zp(const float* __restrict__ z, const float* __restrict__ lnzs,
                     const float* __restrict__ lnzb, const float* __restrict__ Wz,
                     float* __restrict__ zp) {
  __shared__ float wz[16 * TOK_Z];
  __shared__ float sc[TOK_Z], bi[TOK_Z];
  int t = threadIdx.x;
  for (int i = t; i < 16 * TOK_Z; i += 256) wz[i] = Wz[i];
  if (t < TOK_Z) { sc[t] = lnzs[t]; bi[t] = lnzb[t]; }
  __syncthreads();
  int wv = t >> 5, lane = t & 31;
  int pair = blockIdx.x * 8 + wv;                 // 0..8191
  int k = pair >> 6, r = pair & 63, tql = r >> 4, tkl = r & 15;
  int tq = 4 * k + tql;
  int tk = 4 * k - 6 + tkl;
  float prt[16];
  bool valid = (unsigned)tk < (unsigned)T_TOK;
  if (valid) {
    const float4 x4 = reinterpret_cast<const float4*>(z + (size_t)(tq * T_TOK + tk) * TOK_Z)[lane];
    float s1 = x4.x + x4.y + x4.z + x4.w;
    float s2 = x4.x * x4.x + x4.y * x4.y + x4.z * x4.z + x4.w * x4.w;
    for (int off = 16; off; off >>= 1) { s1 += __shfl_xor(s1, off, 32); s2 += __shfl_xor(s2, off, 32); }
    float mu = s1 * (1.f / TOK_Z);
    float rs = rsqrtf(s2 * (1.f / TOK_Z) - mu * mu + 1e-5f);
    int c0 = lane * 4;
    float l0 = (x4.x - mu) * rs * sc[c0]     + bi[c0];
    float l1 = (x4.y - mu) * rs * sc[c0 + 1] + bi[c0 + 1];
    float l2 = (x4.z - mu) * rs * sc[c0 + 2] + bi[c0 + 2];
    float l3 = (x4.w - mu) * rs * sc[c0 + 3] + bi[c0 + 3];
#pragma unroll
    for (int j = 0; j < 16; j++) {
      const float* w = wz + j * TOK_Z + c0;
      prt[j] = l0 * w[0] + l1 * w[1] + l2 * w[2] + l3 * w[3];
    }
#pragma unroll
    for (int j = 0; j < 16; j++)
      for (int off = 16; off; off >>= 1) prt[j] += __shfl_xor(prt[j], off, 32);
  } else {
#pragma unroll
    for (int j = 0; j < 16; j++) prt[j] = 0.f;
  }
  if (lane == 0) {
#pragma unroll
    for (int j = 0; j < 16; j++) zp[pair * 16 + j] = prt[j];
  }
}

// ---------------------------------------------------------------- kernel D
// pq = relu(c)@Wq^T, pk = relu(c)@Wk^T   [4096 x 128] x [128 x 16] (x2)
__global__ void k_pq_pk(const float* __restrict__ cv, const float* __restrict__ Wq,
                        const float* __restrict__ Wk, float* __restrict__ pq,
                        float* __restrict__ pk) {
  int wv   = (blockIdx.x * blockDim.x + threadIdx.x) >> 5;  // M tile 0..255
  int lane = threadIdx.x & 31;
  int n16  = lane & 15;
  int half = lane >> 4;
  int m    = wv * 16 + n16;
  v8f aq = {}, ak = {};
  for (int k0 = 0; k0 < ATOM_S; k0 += 4) {
    int kk = k0 + 2 * half;
    v2f a;
    a.x = fmaxf(cv[m * ATOM_S + kk], 0.f);
    a.y = fmaxf(cv[m * ATOM_S + kk + 1], 0.f);
    v2f bq, bk2;
    bq.x  = Wq[n16 * ATOM_S + kk];  bq.y  = Wq[n16 * ATOM_S + kk + 1];
    bk2.x = Wk[n16 * ATOM_S + kk];  bk2.y = Wk[n16 * ATOM_S + kk + 1];
    aq = WMMA_F32(a, bq, aq);
    ak = WMMA_F32(a, bk2, ak);
  }
#pragma unroll
  for (int i = 0; i < 8; i++) {
    int row = wv * 16 + i + 8 * half;
    pq[row * 16 + n16] = aq[i];
    pk[row * 16 + n16] = ak[i];
  }
}

// ---------------------------------------------------------------- kernel E
// Fused p: base assembly + gathers + 3-layer WMMA MLP + residual.
// One block = one (window k, query w); 128 threads = 128 key rows.
// MLP weights are staged LDS-async (ASYNCcnt) and awaited only after the p16
// assembly phase, overlapping the fetch with the latency-bound gather work.
constexpr int RP = 130;  // row pitch for LDS tiles
__global__ void k_pair_fused(const float* __restrict__ pos, const unsigned char* __restrict__ pad,
                             const int* __restrict__ uid,
                             const float* __restrict__ Wp, const float* __restrict__ Wd,
                             const float* __restrict__ Wm,
                             const float* __restrict__ W1, const float* __restrict__ W2,
                             const float* __restrict__ W3,
                             const float* __restrict__ zp, const float* __restrict__ pq,
                             const float* __restrict__ pk, float* __restrict__ p_out) {
  __shared__ float tA[16 * RP];
  __shared__ float tB[16 * RP];
  __shared__ float wp[48], wd[16], wm[16], w1[256], w2[256], w3[256], zs[256], pqv[16];
  int t  = threadIdx.x;
  int bk = blockIdx.x >> 5;     // window 0..127
  int w  = blockIdx.x & 31;     // query 0..31
  int q_atom = bk * 32 + w;
  int k_atom = bk * 32 - 48 + t;
  bool in = (unsigned)k_atom < (unsigned)N_ATOM;

  // warm the gather lines early (global_prefetch_b8); consumed after barrier
  if (in) {
    __builtin_prefetch(&pk[k_atom * 16], 0, 3);
    __builtin_prefetch(&pos[k_atom * 3], 0, 3);
  }

#if USE_ASYNC_LDS
  // MLP weights not needed until after p16 assembly: stage via async-to-LDS.
  if (t < 64) {
    typedef int v4i_vs __attribute__((vector_size(4 * sizeof(int))));
    __builtin_amdgcn_global_load_async_to_lds_b128((v4i_vs*)(W1 + t * 4), (v4i_vs*)&w1[t * 4], 0, 0);
    __builtin_amdgcn_global_load_async_to_lds_b128((v4i_vs*)(W2 + t * 4), (v4i_vs*)&w2[t * 4], 0, 0);
    __builtin_amdgcn_global_load_async_to_lds_b128((v4i_vs*)(W3 + t * 4), (v4i_vs*)&w3[t * 4], 0, 0);
  }
#else
  for (int i = t; i < 256; i += 128) { w1[i] = W1[i]; w2[i] = W2[i]; w3[i] = W3[i]; }
#endif
  if (t < 48) wp[t] = Wp[t];
  if (t < 16) { wd[t] = Wd[t]; wm[t] = Wm[t]; pqv[t] = pq[q_atom * 16 + t]; }
  {
    int zbase = (bk * 4 + (w >> 3)) * 256;
    for (int i = t; i < 256; i += 128) zs[i] = zp[zbase + i];
  }
  __syncthreads();

  float pq0 = pos[q_atom * 3], pq1 = pos[q_atom * 3 + 1], pq2 = pos[q_atom * 3 + 2];
  float d0 = (in ? pos[k_atom * 3]     : 0.f) - pq0;
  float d1 = (in ? pos[k_atom * 3 + 1] : 0.f) - pq1;
  float d2 = (in ? pos[k_atom * 3 + 2] : 0.f) - pq2;
  float dn = 1.f / (1.f + d0 * d0 + d1 * d1 + d2 * d2);
  int   uq = uid[q_atom];
  float vf = (in && pad[q_atom] && pad[k_atom] && uid[k_atom] == uq) ? 1.f : 0.f;

  float p16[16];
#pragma unroll
  for (int j = 0; j < 16; j++) {
    float base = vf * (wp[j * 3] * d0 + wp[j * 3 + 1] * d1 + wp[j * 3 + 2] * d2 +
                       wd[j] * dn + wm[j] * vf);
    float extra = pqv[j];
    if (in) extra += zs[(t >> 3) * 16 + j] + pk[k_atom * 16 + j];
    p16[j] = base + extra;
    tA[j * RP + t] = fmaxf(p16[j], 0.f);  // relu(p) in [ch][row] layout for WMMA A
  }
#if USE_ASYNC_LDS
  __builtin_amdgcn_s_wait_asynccnt(0);    // w1/w2/w3 landed in LDS
#endif
  __syncthreads();

  int lane = t & 31, wv = t >> 5;
  int n16 = lane & 15, half = lane >> 4;
  auto layer = [&](const float* src, float* dst, const float* wgt, bool relu) {
#pragma unroll
    for (int tt = 0; tt < 2; ++tt) {
      int tile = wv * 2 + tt;                    // 4 waves x 2 tiles = 8 x 16 rows
      v8f acc = {};
#pragma unroll
      for (int ks = 0; ks < 4; ++ks) {
        int kk  = ks * 4 + 2 * half;
        int row = tile * 16 + n16;
        v2f a, b;
        a.x = src[kk * RP + row];
        a.y = src[(kk + 1) * RP + row];
        b.x = wgt[n16 * 16 + kk];                // B[k][n] = W[n][k]
        b.y = wgt[n16 * 16 + kk + 1];
        acc = WMMA_F32(a, b, acc);
      }
#pragma unroll
      for (int i = 0; i < 8; i++) {
        int row = tile * 16 + i + 8 * half;
        float v = acc[i];
        dst[n16 * RP + row] = relu ? fmaxf(v, 0.f) : v;
      }
    }
  };
  layer(tA, tB, w1, true);  __syncthreads();
  layer(tB, tA, w2, true);  __syncthreads();
  layer(tA, tB, w3, false); __syncthreads();

  float* out = p_out + (size_t)(bk * 4096 + w * 128 + t) * 16;
#pragma unroll
  for (int j = 0; j < 16; j++) out[j] = p16[j] + tB[j * RP + t];
}

// ---------------------------------------------------------------- launcher
extern "C" void kernel_launch(void* const* d_in, const int* in_sizes, int n_in,
                              void* d_out, int out_size, void* d_ws, size_t ws_size,
                              hipStream_t stream) {
  (void)in_sizes; (void)n_in; (void)out_size; (void)ws_size;
  const float* ref_pos   = (const float*)d_in[0];
  const float* ref_chg   = (const float*)d_in[1];
  const float* ref_elem  = (const float*)d_in[2];
  const float* ref_chars = (const float*)d_in[3];
  // d_in[4] atom_to_token: one-hot(ref_space_uid) -> handled as gathers via uid
  const float* s_trunk   = (const float*)d_in[5];
  const float* z         = (const float*)d_in[6];
  const float* W_feat    = (const float*)d_in[7];
  const float* b_feat    = (const float*)d_in[8];
  const float* W_rpos    = (const float*)d_in[9];
  const float* W_rdist   = (const float*)d_in[10];
  const float* W_maskw   = (const float*)d_in[11];
  const float* ln_s_sc   = (const float*)d_in[12];
  const float* ln_s_bi   = (const float*)d_in[13];
  const float* W_s2c     = (const float*)d_in[14];
  const float* ln_z_sc   = (const float*)d_in[15];
  const float* ln_z_bi   = (const float*)d_in[16];
  const float* W_z2p     = (const float*)d_in[17];
  const float* W_cpq     = (const float*)d_in[18];
  const float* W_cpk     = (const float*)d_in[19];
  const float* W_mlp1    = (const float*)d_in[20];
  const float* W_mlp2    = (const float*)d_in[21];
  const float* W_mlp3    = (const float*)d_in[22];
  const unsigned char* pad = (const unsigned char*)d_in[23];
  const int* uid         = (const int*)d_in[24];

  float* q_out = (float*)d_out;
  float* c_out = q_out + N_ATOM * ATOM_S;
  float* p_out = c_out + N_ATOM * ATOM_S;

  float* ws     = (float*)d_ws;
  float* s_mean = ws + WS_SMEAN;
  float* s_rstd = ws + WS_SRSTD;
  float* s2c    = ws + WS_S2C;
  float* zp     = ws + WS_ZP;
  float* pqs    = ws + WS_PQ;
  float* pks    = ws + WS_PK;

  k_atom_embed<<<32, 256, 0, stream>>>(ref_pos, ref_chg, ref_elem, ref_chars,
                                       W_feat, b_feat, q_out);
  k_s_stats<<<64, 256, 0, stream>>>(s_trunk, s_mean, s_rstd);
  k_s_to_c<<<4, 256, 0, stream>>>(s_trunk, s_mean, s_rstd, ln_s_sc, ln_s_bi, W_s2c, s2c);
  k_c_add<<<2048, 256, 0, stream>>>(q_out, s2c, uid, c_out);
  k_zp<<<1024, 256, 0, stream>>>(z, ln_z_sc, ln_z_bi, W_z2p, zp);
  k_pq_pk<<<32, 256, 0, stream>>>(c_out, W_cpq, W_cpk, pqs, pks);
  k_pair_fused<<<4096, 128, 0, stream>>>(ref_pos, pad, uid, W_rpos, W_rdist, W_maskw,
                                         W_mlp1, W_mlp2, W_mlp3, zp, pqs, pks, p_out);
}